// NADE_29051158790368
// MI455X (gfx1250) — compile-verified
//
#include <hip/hip_runtime.h>
#include <hip/hip_bf16.h>

// NADE forward for MI455X (gfx1250, wave32).
// B=512, D=1024, H=512, C=4.
//   grid = 32 batch tiles (16 rows) x 32 d-chunks (32 steps) = 1024 blocks, 256 thr (8 waves).
//   Phase A: a_start = c + x[:, :d0] @ W[:, :d0]^T  via V_WMMA_F32_16X16X4_F32 (pure fp32).
//   Phase B: 32 serial steps with a (16x512 f32) in LDS; per-step 16x512x4 GEMM via fp32
//            WMMA (K split across 8 waves), log-softmax over C=4 with wave32 __shfl_xor,
//            rank-1 a-update. V_d/W_d/x_d for step s+1 are prefetched with
//            GLOBAL_LOAD_ASYNC_TO_LDS (ASYNCcnt) into double buffers while step s computes.

#define Bsz 512
#define Dd  1024
#define Hh  512
#define Cc  4
#define CHUNK 32
#define NCHUNK (Dd / CHUNK)   // 32
#define BT 16                 // batch tile rows
#define NBT (Bsz / BT)        // 32

typedef float v2f __attribute__((ext_vector_type(2)));
typedef float v8f __attribute__((ext_vector_type(8)));
typedef int   v4i __attribute__((ext_vector_type(4)));

#if defined(__AMDGCN__) && \
    __has_builtin(__builtin_amdgcn_global_load_async_to_lds_b128) && \
    __has_builtin(__builtin_amdgcn_global_load_async_to_lds_b32) && \
    __has_builtin(__builtin_amdgcn_s_wait_asynccnt)
#define HAVE_ASYNC_LDS 1
#else
#define HAVE_ASYNC_LDS 0
#endif

#if HAVE_ASYNC_LDS
// Exact parameter types per clang diagnostic: AS(1) int4* / int* sources,
// matching AS(3) destinations. Casts via uintptr_t (generic LDS address low
// 32 bits are the LDS byte offset per ISA 10.2 aperture rules).
typedef __attribute__((address_space(1))) v4i* g4_t;
typedef __attribute__((address_space(1))) int* g1_t;
typedef __attribute__((address_space(3))) v4i* l4_t;
typedef __attribute__((address_space(3))) int* l1_t;

__device__ __forceinline__ void async_cp_b128(const float* gsrc, float* ldst) {
    __builtin_amdgcn_global_load_async_to_lds_b128(
        (g4_t)(uintptr_t)gsrc, (l4_t)(uintptr_t)ldst, 0, 0);
}
__device__ __forceinline__ void async_cp_b32(const float* gsrc, float* ldst) {
    __builtin_amdgcn_global_load_async_to_lds_b32(
        (g1_t)(uintptr_t)gsrc, (l1_t)(uintptr_t)ldst, 0, 0);
}
#endif

__device__ __forceinline__ float sigmoidf_fast(float v) {
    return 1.0f / (1.0f + __expf(-v));
}

// Stage V_d (512x4 f32), W[:,d] (512 f32), x[tile,d] (16 f32) into the parity buffers.
__device__ __forceinline__ void stage_step(const float* __restrict__ V,
                                           const float* __restrict__ W,
                                           const float* __restrict__ x,
                                           int tid, int rb, int d,
                                           float* vb, float* wb, float* xb)
{
    const int h  = tid;        // 0..255
    const int h2 = tid + 256;  // 256..511
#if HAVE_ASYNC_LDS
    async_cp_b128(V + ((size_t)h * Dd + d) * Cc,  vb + h * Cc);
    async_cp_b128(V + ((size_t)h2 * Dd + d) * Cc, vb + h2 * Cc);
    async_cp_b32(W + (size_t)h * Dd + d,  wb + h);
    async_cp_b32(W + (size_t)h2 * Dd + d, wb + h2);
    if (tid < BT) {
        async_cp_b32(x + (size_t)(rb + tid) * Dd + d, xb + tid);
    }
#else
    *(float4*)&vb[h * Cc]  = *(const float4*)(V + ((size_t)h * Dd + d) * Cc);
    *(float4*)&vb[h2 * Cc] = *(const float4*)(V + ((size_t)h2 * Dd + d) * Cc);
    wb[h]  = W[(size_t)h * Dd + d];
    wb[h2] = W[(size_t)h2 * Dd + d];
    if (tid < BT) xb[tid] = x[(size_t)(rb + tid) * Dd + d];
#endif
}

__global__ __launch_bounds__(256)
void nade_fwd_kernel(const float* __restrict__ x,   // [B][D]
                     const float* __restrict__ V,   // [H][D][C]
                     const float* __restrict__ bb,  // [D][C]
                     const float* __restrict__ W,   // [H][D]
                     const float* __restrict__ cc,  // [H]
                     float* __restrict__ out)       // y_hat [B][D*C] ++ p_hat [B][D*C]
{
    __shared__ float a_lds[Hh * BT];        // a, h-major: a_lds[h*16 + m]    (32 KB)
    __shared__ float v_buf[2][Hh * Cc];     // V_d double buffer              (16 KB)
    __shared__ float w_buf[2][Hh];          // W[:, d] double buffer          (4 KB)
    __shared__ float xs_buf[2][BT];         // x[tile rows, d] double buffer
    __shared__ float red[8 * BT * Cc];      // per-wave partial logits        (2 KB)

    const int tid  = threadIdx.x;
    const int lane = tid & 31;
    const int wave = tid >> 5;
    const int lm   = lane & 15;             // M/N index within a 16-group
    const int lh   = lane >> 4;             // 0: K=k0,k0+1   1: K=k0+2,k0+3

    const int tile  = blockIdx.x / NCHUNK;  // batch tile 0..31
    const int chunk = blockIdx.x % NCHUNK;  // d-chunk    0..31
    const int rb    = tile * BT;            // batch row base
    const int d0    = chunk * CHUNK;        // first d of this chunk

    // ---------------- Phase A: checkpoint a = c + x[:, :d0] @ W[:, :d0]^T ----------------
    v8f acc[4];
    #pragma unroll
    for (int i = 0; i < 4; ++i) acc[i] = (v8f){0.f,0.f,0.f,0.f,0.f,0.f,0.f,0.f};
    {
        const float* xrow  = x + (size_t)(rb + lm) * Dd;
        const float* wrow0 = W + (size_t)((wave * 4 + 0) * 16 + lm) * Dd;
        const float* wrow1 = W + (size_t)((wave * 4 + 1) * 16 + lm) * Dd;
        const float* wrow2 = W + (size_t)((wave * 4 + 2) * 16 + lm) * Dd;
        const float* wrow3 = W + (size_t)((wave * 4 + 3) * 16 + lm) * Dd;
        for (int k0 = 0; k0 < d0; k0 += 4) {
            const int ka = k0 + lh * 2;
            v2f A  = *(const v2f*)(xrow + ka);
            v2f B0 = *(const v2f*)(wrow0 + ka);
            v2f B1 = *(const v2f*)(wrow1 + ka);
            v2f B2 = *(const v2f*)(wrow2 + ka);
            v2f B3 = *(const v2f*)(wrow3 + ka);
            acc[0] = __builtin_amdgcn_wmma_f32_16x16x4_f32(false, A, false, B0, (short)0, acc[0], false, false);
            acc[1] = __builtin_amdgcn_wmma_f32_16x16x4_f32(false, A, false, B1, (short)0, acc[1], false, false);
            acc[2] = __builtin_amdgcn_wmma_f32_16x16x4_f32(false, A, false, B2, (short)0, acc[2], false, false);
            acc[3] = __builtin_amdgcn_wmma_f32_16x16x4_f32(false, A, false, B3, (short)0, acc[3], false, false);
        }
    }
    #pragma unroll
    for (int i = 0; i < 4; ++i) {
        const int h = (wave * 4 + i) * 16 + lm;   // C/D: N = lane%16
        const float cv = cc[h];
        #pragma unroll
        for (int r = 0; r < 8; ++r) {
            const int m = r + 8 * lh;             // C/D: M = r + 8*(lane>=16)
            a_lds[h * BT + m] = acc[i][r] + cv;
        }
    }
    __syncthreads();

    // ---------------- Phase B: 32 serial steps ----------------
#if HAVE_ASYNC_LDS
    // Prologue: async-stage step 0's operands.
    stage_step(V, W, x, tid, rb, d0, v_buf[0], w_buf[0], xs_buf[0]);
#endif
    for (int s = 0; s < CHUNK; ++s) {
        const int d = d0 + s;
        float* vb = v_buf[s & 1];
        float* wb = w_buf[s & 1];
        float* xb = xs_buf[s & 1];

#if HAVE_ASYNC_LDS
        __builtin_amdgcn_s_wait_asynccnt(0);   // own async copies landed in LDS
        __syncthreads();                       // everyone's copies visible; prev update done
        if (s + 1 < CHUNK) {
            stage_step(V, W, x, tid, rb, d + 1,
                       v_buf[(s + 1) & 1], w_buf[(s + 1) & 1], xs_buf[(s + 1) & 1]);
        }
#else
        stage_step(V, W, x, tid, rb, d, vb, wb, xb);
        __syncthreads();
#endif

        // logits = sigmoid(a) @ V_d : each wave takes K in [wave*64, wave*64+64).
        v8f c8 = (v8f){0.f,0.f,0.f,0.f,0.f,0.f,0.f,0.f};
        const int kbase = wave * 64;
        #pragma unroll
        for (int kk = 0; kk < 16; ++kk) {
            const int k0 = kbase + kk * 4 + lh * 2;
            const float a0 = a_lds[k0 * BT + lm];
            const float a1 = a_lds[(k0 + 1) * BT + lm];
            v2f A;
            A.x = sigmoidf_fast(a0);
            A.y = sigmoidf_fast(a1);
            v2f Bv;
            Bv.x = (lm < Cc) ? vb[k0 * Cc + lm]       : 0.0f;
            Bv.y = (lm < Cc) ? vb[(k0 + 1) * Cc + lm] : 0.0f;
            c8 = __builtin_amdgcn_wmma_f32_16x16x4_f32(false, A, false, Bv, (short)0, c8, false, false);
        }
        if (lm < Cc) {
            #pragma unroll
            for (int r = 0; r < 8; ++r) {
                const int m = r + 8 * lh;
                red[(wave * BT + m) * Cc + lm] = c8[r];
            }
        }
        __syncthreads();

        // Reduce partials, bias, log-softmax over C=4, write both outputs.
        if (tid < BT * Cc) {
            const int m = tid >> 2;
            const int c = tid & 3;
            float lg = bb[d * Cc + c];
            #pragma unroll
            for (int w = 0; w < 8; ++w) lg += red[(w * BT + m) * Cc + c];
            float mx = lg;
            mx = fmaxf(mx, __shfl_xor(mx, 1, 32));
            mx = fmaxf(mx, __shfl_xor(mx, 2, 32));
            float e = __expf(lg - mx);
            float se = e;
            se += __shfl_xor(se, 1, 32);
            se += __shfl_xor(se, 2, 32);
            const float lse = mx + __logf(se);
            const size_t o = (size_t)(rb + m) * (Dd * Cc) + (size_t)d * Cc + c;
            out[o] = lg;                                  // y_hat
            out[(size_t)Bsz * Dd * Cc + o] = lg - lse;    // p_hat
        }

        // Rank-1 update: a[h][m] += x[m,d] * W[h,d]. (reads parity-s buffers; next stage
        // writes the other parity, so no trailing barrier needed)
        {
            const int m  = tid & 15;
            const int hb = (tid >> 4) * 32;
            const float xm = xb[m];
            #pragma unroll 8
            for (int q = 0; q < 32; ++q) {
                const int h = hb + q;
                a_lds[h * BT + m] += xm * wb[h];
            }
        }
    }
}

extern "C" void kernel_launch(void* const* d_in, const int* in_sizes, int n_in,
                              void* d_out, int out_size, void* d_ws, size_t ws_size,
                              hipStream_t stream) {
    (void)in_sizes; (void)n_in; (void)out_size; (void)d_ws; (void)ws_size;
    const float* x  = (const float*)d_in[0];
    const float* V  = (const float*)d_in[1];
    const float* bb = (const float*)d_in[2];
    const float* W  = (const float*)d_in[3];
    const float* cc = (const float*)d_in[4];
    float* out = (float*)d_out;
    dim3 grid(NBT * NCHUNK);   // 1024 blocks
    dim3 block(256);           // 8 wave32
    nade_fwd_kernel<<<grid, block, 0, stream>>>(x, V, bb, W, cc, out);
}